// loss_separation_42288247997088
// MI455X (gfx1250) — compile-verified
//
#include <hip/hip_runtime.h>

typedef __attribute__((ext_vector_type(2))) float v2f;
typedef __attribute__((ext_vector_type(8))) float v8f;

#define NKP    8192
#define NB     64
#define DDIM   128                         // NB * 2
#define TILE   128
#define LSTR   132                         // LDS row stride (floats); 132 % 8 == 4 -> conflict-free frag loads
#define NTILES (NKP / TILE)                // 64
#define NBLOCKS (NTILES * (NTILES + 1) / 2) // 2080 (upper-triangular tile grid)

__global__ void zero_out_kernel(float* out) { out[0] = 0.0f; }

__launch_bounds__(256, 1)
__global__ void pair_loss_kernel(const float* __restrict__ kp, float* __restrict__ out)
{
    __shared__ float As[TILE * LSTR];
    __shared__ float Bs[TILE * LSTR];
    __shared__ float sqA[TILE];
    __shared__ float sqB[TILE];
    __shared__ float wsum[8];

    const int tid  = threadIdx.x;
    const int lane = tid & 31;   // wave32
    const int wave = tid >> 5;

    // Triangular tile index -> (by, bx) with bx >= by (exploit symmetry of G).
    int rem = blockIdx.x;
    int by = 0;
    while (rem >= NTILES - by) { rem -= NTILES - by; ++by; }
    const int bx   = by + rem;
    const int row0 = by * TILE;
    const int col0 = bx * TILE;
    const float scale = (bx == by) ? 1.0f : 2.0f;

    // ---- stage X tiles into LDS --------------------------------------
    // X[n][d], d = 2*b + c  maps to kp[b*NKP*2 + n*2 + c]; one float2 load
    // at kp2[b*NKP + n] yields X[n][2b..2b+1]. Coalesced over n.
    const float2* kp2 = (const float2*)kp;
    for (int it = 0; it < (TILE * NB) / 256; ++it) {
        int idx = it * 256 + tid;          // 0 .. 8191
        int nl  = idx & (TILE - 1);        // local row (consecutive across lanes)
        int b   = idx >> 7;                // batch index 0..63
        float2 a  = kp2[b * NKP + row0 + nl];
        As[nl * LSTR + 2 * b]     = a.x;
        As[nl * LSTR + 2 * b + 1] = a.y;
        float2 bb = kp2[b * NKP + col0 + nl];
        Bs[nl * LSTR + 2 * b]     = bb.x;
        Bs[nl * LSTR + 2 * b + 1] = bb.y;
    }
    __syncthreads();

    // ---- per-row squared norms ---------------------------------------
    if (tid < TILE) {
        float s = 0.f;
        for (int d = 0; d < DDIM; ++d) { float v = As[tid * LSTR + d]; s += v * v; }
        sqA[tid] = s;
    } else {
        int r = tid - TILE;
        float s = 0.f;
        for (int d = 0; d < DDIM; ++d) { float v = Bs[r * LSTR + d]; s += v * v; }
        sqB[r] = s;
    }
    __syncthreads();

    // ---- f32 WMMA GEMM: G_tile = Xa . Xb^T ---------------------------
    // Each wave: 32x64 slab = 2x4 tiles of 16x16, K chained in steps of 4.
    const int wm    = wave >> 1;           // 0..3 : 32-row slab
    const int wn    = wave & 1;            // 0..1 : 64-col slab
    const int mbase = wm * 32;
    const int nbase = wn * 64;
    const int lrow  = lane & 15;
    const int khalf = (lane >> 4) * 2;     // A/B frag: half-wave K sub-offset

    v8f acc[2][4] = {};

#pragma unroll 4
    for (int kk = 0; kk < DDIM / 4; ++kk) {
        const int kb = kk * 4 + khalf;
        v2f a0 = *(const v2f*)&As[(mbase +      lrow) * LSTR + kb];
        v2f a1 = *(const v2f*)&As[(mbase + 16 + lrow) * LSTR + kb];
        v2f b0 = *(const v2f*)&Bs[(nbase +      lrow) * LSTR + kb];
        v2f b1 = *(const v2f*)&Bs[(nbase + 16 + lrow) * LSTR + kb];
        v2f b2 = *(const v2f*)&Bs[(nbase + 32 + lrow) * LSTR + kb];
        v2f b3 = *(const v2f*)&Bs[(nbase + 48 + lrow) * LSTR + kb];

        acc[0][0] = __builtin_amdgcn_wmma_f32_16x16x4_f32(false, a0, false, b0, (short)0, acc[0][0], false, false);
        acc[0][1] = __builtin_amdgcn_wmma_f32_16x16x4_f32(false, a0, false, b1, (short)0, acc[0][1], false, false);
        acc[0][2] = __builtin_amdgcn_wmma_f32_16x16x4_f32(false, a0, false, b2, (short)0, acc[0][2], false, false);
        acc[0][3] = __builtin_amdgcn_wmma_f32_16x16x4_f32(false, a0, false, b3, (short)0, acc[0][3], false, false);
        acc[1][0] = __builtin_amdgcn_wmma_f32_16x16x4_f32(false, a1, false, b0, (short)0, acc[1][0], false, false);
        acc[1][1] = __builtin_amdgcn_wmma_f32_16x16x4_f32(false, a1, false, b1, (short)0, acc[1][1], false, false);
        acc[1][2] = __builtin_amdgcn_wmma_f32_16x16x4_f32(false, a1, false, b2, (short)0, acc[1][2], false, false);
        acc[1][3] = __builtin_amdgcn_wmma_f32_16x16x4_f32(false, a1, false, b3, (short)0, acc[1][3], false, false);
    }

    // ---- epilogue: d2 -> exp(-0.4*sqrt(d2)), mask diagonal -----------
    // C/D layout: VGPR r holds M=r (lanes 0-15) and M=r+8 (lanes 16-31), N = lane&15.
    float lsum = 0.0f;
    const int rhi = (lane >> 4) << 3;
    for (int mt = 0; mt < 2; ++mt) {
        for (int nt = 0; nt < 4; ++nt) {
            const int jl = nbase + nt * 16 + lrow;
            const int jg = col0 + jl;
            const float sj = sqB[jl];
#pragma unroll
            for (int r = 0; r < 8; ++r) {
                const int il = mbase + mt * 16 + r + rhi;
                const int ig = row0 + il;
                float d2 = sqA[il] + sj - 2.0f * acc[mt][nt][r];
                d2 = fmaxf(d2, 0.0f);
                float e = __expf(-0.4f * __builtin_amdgcn_sqrtf(d2));
                lsum += (ig == jg) ? 0.0f : e;
            }
        }
    }

    // ---- reduction: wave32 shuffle -> LDS -> one atomic per block ----
    for (int off = 16; off > 0; off >>= 1)
        lsum += __shfl_xor(lsum, off, 32);
    if (lane == 0) wsum[wave] = lsum;
    __syncthreads();
    if (tid == 0) {
        float s = 0.f;
        for (int w = 0; w < 8; ++w) s += wsum[w];
        atomicAdd(out, s * scale);
    }
}

extern "C" void kernel_launch(void* const* d_in, const int* in_sizes, int n_in,
                              void* d_out, int out_size, void* d_ws, size_t ws_size,
                              hipStream_t stream) {
    (void)in_sizes; (void)n_in; (void)out_size; (void)d_ws; (void)ws_size;
    const float* kp = (const float*)d_in[0];
    float* out = (float*)d_out;
    zero_out_kernel<<<1, 1, 0, stream>>>(out);
    pair_loss_kernel<<<NBLOCKS, 256, 0, stream>>>(kp, out);
}